// MambaLayer2_CrossAttn_Click_67001489817884
// MI455X (gfx1250) — compile-verified
//
#include <hip/hip_runtime.h>
#include <math.h>

typedef __attribute__((ext_vector_type(2))) float v2f;
typedef __attribute__((ext_vector_type(8))) float v8f;
typedef __attribute__((ext_vector_type(4))) unsigned u32x4;
typedef __attribute__((ext_vector_type(8))) unsigned u32x8;

#define LSEQ   4096
#define NTOK   8192      // BATCH * LSEQ
#define DIMC   320
#define DINNER 640
#define DSTATE 32
#define NHD    8
#define HD     80
#define CONVD  704       // DINNER + 2*DSTATE
#define DPROJ  1352      // 2*DINNER + 2*DSTATE + NHD
#define KCHUNK 160       // K-chunk staged into LDS by the TDM

__device__ __forceinline__ float sigmoidf_(float x) { return 1.f / (1.f + expf(-x)); }

// ---------------------------------------------------------------------------
// Direct fp32 WMMA GEMM (one wave per 16x16 tile): used for tiny M=16 GEMMs.
// C[M,N] = A[M,K] @ W[N,K]^T (+ bias) (+ optional ReLU);  K % 4 == 0.
// ---------------------------------------------------------------------------
__global__ void gemm_wmma_f32(const float* __restrict__ A, const float* __restrict__ W,
                              const float* __restrict__ bias, float* __restrict__ C,
                              int M, int N, int K, int act) {
  const int tn = blockIdx.x, tm = blockIdx.y;
  const int lane = threadIdx.x;
  const int ml = lane & 15, hi = lane >> 4;

  int m = tm * 16 + ml; if (m > M - 1) m = M - 1;
  int n = tn * 16 + ml; if (n > N - 1) n = N - 1;

  const float* pa = A + (size_t)m * K + 2 * hi;
  const float* pw = W + (size_t)n * K + 2 * hi;

  v8f acc = {0.f, 0.f, 0.f, 0.f, 0.f, 0.f, 0.f, 0.f};
  for (int k0 = 0; k0 < K; k0 += 4) {
    __builtin_prefetch(pw + k0 + 128, 0, 0);
    v2f av, bv;
    av.x = pa[k0];  av.y = pa[k0 + 1];
    bv.x = pw[k0];  bv.y = pw[k0 + 1];
    acc = __builtin_amdgcn_wmma_f32_16x16x4_f32(false, av, false, bv,
                                                (short)0, acc, false, false);
  }
  const int nn = tn * 16 + ml;
  if (nn < N) {
    const float bb = bias ? bias[nn] : 0.f;
#pragma unroll
    for (int r = 0; r < 8; ++r) {
      const int mm = tm * 16 + r + 8 * hi;
      if (mm < M) {
        float v = acc[r] + bb;
        if (act == 1) v = fmaxf(v, 0.f);
        C[(size_t)mm * N + nn] = v;
      }
    }
  }
}

// ---------------------------------------------------------------------------
// TDM-staged fp32 WMMA GEMM for the token-sized (M=8192) GEMMs.
// Workgroup = 8 waves covering 128 M-rows x 32 N-cols. The Tensor Data Mover
// DMAs a [rows<=32, clen<=160] slab of W into LDS (tensor_load_to_lds, D# per
// CDNA5 ISA 8.3/8.4); all 8 waves then read B-fragments from LDS (8x reuse).
// Each wave holds two 16x16 f32 accumulators (2 N-subtiles).
// ---------------------------------------------------------------------------
__global__ void gemm_wmma_lds(const float* __restrict__ A, const float* __restrict__ W,
                              const float* __restrict__ bias, float* __restrict__ C,
                              int M, int N, int K) {
  __shared__ float wsm[32 * KCHUNK];
  const int tn = blockIdx.x, tm = blockIdx.y;
  const int wave = threadIdx.x >> 5;
  const int lane = threadIdx.x & 31;
  const int ml = lane & 15, hi = lane >> 4;

  int m = tm * 128 + wave * 16 + ml; if (m > M - 1) m = M - 1;
  const float* pa = A + (size_t)m * K + 2 * hi;
  const unsigned loff = (unsigned)(unsigned long long)(void*)wsm;  // low 32 bits = LDS byte offset

  v8f acc0 = {0.f, 0.f, 0.f, 0.f, 0.f, 0.f, 0.f, 0.f};
  v8f acc1 = {0.f, 0.f, 0.f, 0.f, 0.f, 0.f, 0.f, 0.f};

  int rows = N - tn * 32; if (rows > 32) rows = 32;

  for (int k0 = 0; k0 < K; k0 += KCHUNK) {
    int clen = K - k0; if (clen > KCHUNK) clen = KCHUNK;

    if (threadIdx.x == 0) {
      // ---- Tensor DMA descriptor (D#) ----
      unsigned long long gaddr =
          (unsigned long long)(const void*)(W + (size_t)tn * 32 * K + k0);
      u32x4 g0;
      g0[0] = 1u;                                        // count=1, user mode
      g0[1] = loff;                                      // lds_addr (bytes)
      g0[2] = (unsigned)gaddr;                           // global_addr[31:0]
      g0[3] = ((unsigned)(gaddr >> 32) & 0x01FFFFFFu)    // global_addr[56:32]
              | (2u << 30);                              // type = 2 ("image")
      u32x8 g1;
      g1[0] = 0x00020000u;                               // wg_mask=0, data_size=4B
      g1[1] = ((unsigned)clen & 0xFFFFu) << 16;          // tensor_dim0[15:0]
      g1[2] = ((unsigned)rows << 16);                    // dim0[31:16]=0 | tensor_dim1[15:0]
      g1[3] = ((unsigned)clen << 16);                    // tensor_dim1[31:16]=0 | tile_dim0
      g1[4] = (unsigned)rows;                            // tile_dim1 | tile_dim2=0
      g1[5] = (unsigned)K;                               // tensor_dim0_stride[31:0] (elements)
      g1[6] = 0u;                                        // stride hi | dim1_stride lo
      g1[7] = 0u;
      asm volatile("tensor_load_to_lds %0, %1" :: "s"(g0), "s"(g1) : "memory");
      __builtin_amdgcn_s_wait_tensorcnt(0);
    }
    __syncthreads();

    for (int kk = 0; kk < clen; kk += 4) {
      v2f av, b0, b1;
      av.x = pa[k0 + kk];  av.y = pa[k0 + kk + 1];
      const int kb = kk + 2 * hi;
      b0.x = wsm[ml * clen + kb];         b0.y = wsm[ml * clen + kb + 1];
      b1.x = wsm[(16 + ml) * clen + kb];  b1.y = wsm[(16 + ml) * clen + kb + 1];
      acc0 = __builtin_amdgcn_wmma_f32_16x16x4_f32(false, av, false, b0,
                                                   (short)0, acc0, false, false);
      acc1 = __builtin_amdgcn_wmma_f32_16x16x4_f32(false, av, false, b1,
                                                   (short)0, acc1, false, false);
    }
    __syncthreads();
  }

  const int nnA = tn * 32 + ml;
  const int nnB = tn * 32 + 16 + ml;
  const float bA = (bias && nnA < N) ? bias[nnA] : 0.f;
  const float bB = (bias && nnB < N) ? bias[nnB] : 0.f;
#pragma unroll
  for (int r = 0; r < 8; ++r) {
    const int mm = tm * 128 + wave * 16 + r + 8 * hi;
    if (mm < M) {
      if (nnA < N) C[(size_t)mm * N + nnA] = acc0[r] + bA;
      if (nnB < N) C[(size_t)mm * N + nnB] = acc1[r] + bB;
    }
  }
}

// ---------------------------------------------------------------------------
// LayerNorm over channels of NCHW input, writing token-major (row, C)
// ---------------------------------------------------------------------------
__global__ void ln_in_kernel(const float* __restrict__ x, const float* __restrict__ w,
                             const float* __restrict__ b, float* __restrict__ u) {
  const int t = blockIdx.x & (LSEQ - 1);
  const int bb = blockIdx.x >> 12;
  __shared__ float sb[256];
  float s = 0.f, s2 = 0.f;
  for (int c = threadIdx.x; c < DIMC; c += 256) {
    const float v = x[((size_t)bb * DIMC + c) * LSEQ + t];
    s += v; s2 += v * v;
  }
  sb[threadIdx.x] = s; __syncthreads();
  for (int st = 128; st > 0; st >>= 1) { if (threadIdx.x < st) sb[threadIdx.x] += sb[threadIdx.x + st]; __syncthreads(); }
  const float mean = sb[0] / DIMC; __syncthreads();
  sb[threadIdx.x] = s2; __syncthreads();
  for (int st = 128; st > 0; st >>= 1) { if (threadIdx.x < st) sb[threadIdx.x] += sb[threadIdx.x + st]; __syncthreads(); }
  const float inv = rsqrtf(sb[0] / DIMC - mean * mean + 1e-5f);
  for (int c = threadIdx.x; c < DIMC; c += 256) {
    const float v = x[((size_t)bb * DIMC + c) * LSEQ + t];
    u[(size_t)blockIdx.x * DIMC + c] = (v - mean) * inv * w[c] + b[c];
  }
}

// ---------------------------------------------------------------------------
// Generic LayerNorm / RMSNorm with optional residual input. out may alias in.
// ---------------------------------------------------------------------------
__global__ void norm_kernel(const float* __restrict__ in, const float* __restrict__ res,
                            const float* __restrict__ w, const float* __restrict__ b,
                            float* __restrict__ out, int width, int rms) {
  const size_t row = blockIdx.x;
  const float* pi = in + row * width;
  const float* pr = res ? res + row * width : nullptr;
  __shared__ float sb[256];
  float s = 0.f, s2 = 0.f;
  for (int c = threadIdx.x; c < width; c += 256) {
    const float v = pi[c] + (pr ? pr[c] : 0.f);
    s += v; s2 += v * v;
  }
  sb[threadIdx.x] = s; __syncthreads();
  for (int st = 128; st > 0; st >>= 1) { if (threadIdx.x < st) sb[threadIdx.x] += sb[threadIdx.x + st]; __syncthreads(); }
  const float mean = sb[0] / width; __syncthreads();
  sb[threadIdx.x] = s2; __syncthreads();
  for (int st = 128; st > 0; st >>= 1) { if (threadIdx.x < st) sb[threadIdx.x] += sb[threadIdx.x + st]; __syncthreads(); }
  const float ms = sb[0] / width;
  const float mu = rms ? 0.f : mean;
  const float var = rms ? ms : ms - mean * mean;
  const float inv = rsqrtf(var + 1e-5f);
  for (int c = threadIdx.x; c < width; c += 256) {
    const float v = pi[c] + (pr ? pr[c] : 0.f);
    out[row * width + c] = (v - mu) * inv * w[c] + (b ? b[c] : 0.f);
  }
}

// dt = softplus(zx[..., -8:] + dt_bias);  dA = exp(-exp(A_log) * dt)
__global__ void dtda_kernel(const float* __restrict__ zx, const float* __restrict__ dt_bias,
                            const float* __restrict__ A_log, float* __restrict__ dt,
                            float* __restrict__ dA) {
  const int idx = blockIdx.x * blockDim.x + threadIdx.x;
  if (idx >= NTOK * NHD) return;
  const int row = idx >> 3, h = idx & 7;
  const float raw = zx[(size_t)row * DPROJ + (DINNER + CONVD) + h] + dt_bias[h];
  const float dv = (raw > 20.f) ? raw : log1pf(expf(raw));
  dt[idx] = dv;
  dA[idx] = expf(-expf(A_log[h]) * dv);
}

// Causal depthwise conv (width 4) over xBC slice + bias + SiLU
__global__ void conv_silu_kernel(const float* __restrict__ zx, const float* __restrict__ cw,
                                 const float* __restrict__ cb, float* __restrict__ xbc) {
  const int idx = blockIdx.x * blockDim.x + threadIdx.x;
  if (idx >= NTOK * CONVD) return;
  const int row = idx / CONVD, c = idx - row * CONVD;
  const int bI = row >> 12, l = row & (LSEQ - 1);
  float acc = cb[c];
#pragma unroll
  for (int j = 0; j < 4; ++j) {
    const int ls = l - 3 + j;
    if (ls >= 0) acc += cw[c * 4 + j] * zx[((size_t)(bI * LSEQ + ls)) * DPROJ + DINNER + c];
  }
  xbc[(size_t)row * CONVD + c] = acc * sigmoidf_(acc);
}

// Sequential selective scan: one block per (batch, head); thread p holds h[p][0..31]
__global__ void scan_kernel(const float* __restrict__ xbc, const float* __restrict__ dt,
                            const float* __restrict__ dA, const float* __restrict__ Dp,
                            float* __restrict__ y) {
  const int bI = blockIdx.x >> 3, h = blockIdx.x & 7;
  const int tid = threadIdx.x;
  __shared__ float Bs[DSTATE], Cs[DSTATE], sc[2];
  float hreg[DSTATE];
#pragma unroll
  for (int n = 0; n < DSTATE; ++n) hreg[n] = 0.f;
  const float Dv = Dp[h];
  for (int l = 0; l < LSEQ; ++l) {
    const size_t row = (size_t)bI * LSEQ + l;
    if (tid < DSTATE)            Bs[tid] = xbc[row * CONVD + DINNER + tid];
    else if (tid < 2 * DSTATE)   Cs[tid - DSTATE] = xbc[row * CONVD + DINNER + DSTATE + (tid - DSTATE)];
    if (tid == 64) { sc[0] = dt[row * NHD + h]; sc[1] = dA[row * NHD + h]; }
    __syncthreads();
    if (tid < HD) {
      const float xv = xbc[row * CONVD + h * HD + tid];
      const float dtx = sc[0] * xv;
      const float dAv = sc[1];
      float acc = 0.f;
#pragma unroll
      for (int n = 0; n < DSTATE; ++n) { hreg[n] = hreg[n] * dAv + dtx * Bs[n]; acc += hreg[n] * Cs[n]; }
      y[row * DINNER + h * HD + tid] = acc + Dv * xv;
    }
    __syncthreads();
  }
}

// y *= silu(z)   (z = first DINNER columns of zx)
__global__ void gate_kernel(const float* __restrict__ zx, float* __restrict__ y) {
  const size_t idx = (size_t)blockIdx.x * blockDim.x + threadIdx.x;
  if (idx >= (size_t)NTOK * DINNER) return;
  const size_t row = idx / DINNER; const int d = (int)(idx - row * DINNER);
  const float z = zx[row * DPROJ + d];
  y[idx] = y[idx] * (z * sigmoidf_(z));
}

// Fourier positional encoding for the dense 16x16x16 grid
__global__ void kpe_kernel(const float* __restrict__ G, float* __restrict__ kpe) {
  const int idx = blockIdx.x * blockDim.x + threadIdx.x;
  if (idx >= LSEQ * 160) return;
  const int t = idx / 160, j = idx - t * 160;
  const float g0 = ((t >> 8) + 0.5f) * 0.125f - 1.f;
  const float g1 = (((t >> 4) & 15) + 0.5f) * 0.125f - 1.f;
  const float g2 = ((t & 15) + 0.5f) * 0.125f - 1.f;
  float c = g0 * G[j] + g1 * G[160 + j] + g2 * G[320 + j];
  c *= 6.283185307179586f;
  kpe[(size_t)t * DIMC + j]        = sinf(c);
  kpe[(size_t)t * DIMC + 160 + j]  = cosf(c);
}

// Query positional encoding + point embedding lookup (16 points)
__global__ void qpe_kernel(const float* __restrict__ coords, const int* __restrict__ labels,
                           const float* __restrict__ G, const float* __restrict__ pemb,
                           float* __restrict__ qpe) {
  const int idx = blockIdx.x * blockDim.x + threadIdx.x;
  if (idx >= 16 * 160) return;
  const int pt = idx / 160, j = idx - pt * 160;
  float c = 0.f;
#pragma unroll
  for (int d = 0; d < 3; ++d) {
    const float pc = (coords[pt * 3 + d] + 0.5f) * (1.f / 16.f);
    c += (2.f * pc - 1.f) * G[d * 160 + j];
  }
  c *= 6.283185307179586f;
  const int lab = labels[pt];
  qpe[pt * DIMC + j]       = sinf(c) + pemb[lab * DIMC + j];
  qpe[pt * DIMC + 160 + j] = cosf(c) + pemb[lab * DIMC + 160 + j];
}

// out[i] = a[i] + pe[i % psz]
__global__ void add_bcast_kernel(const float* __restrict__ a, const float* __restrict__ pe,
                                 float* __restrict__ o, size_t n, size_t psz) {
  const size_t i = (size_t)blockIdx.x * blockDim.x + threadIdx.x;
  if (i < n) o[i] = a[i] + pe[i % psz];
}

// Online-softmax attention: one thread per (batch, head, query)
__global__ void attn_kernel(const float* __restrict__ q, const float* __restrict__ k,
                            const float* __restrict__ v, float* __restrict__ out,
                            int nq, int nk, int nh, int dh) {
  const int idx = blockIdx.x * blockDim.x + threadIdx.x;
  const int total = 2 * nh * nq;
  if (idx >= total) return;
  const int b = idx / (nh * nq);
  const int rem = idx - b * nh * nq;
  const int h = rem / nq, qi = rem - h * nq;
  const int di = nh * dh;
  const float* qp = q + ((size_t)(b * nq + qi)) * di + h * dh;
  float qreg[HD], acc[HD];
  for (int d = 0; d < dh; ++d) { qreg[d] = qp[d]; acc[d] = 0.f; }
  float mrun = -3.0e38f, lsum = 0.f;
  const float scale = rsqrtf((float)dh);
  for (int kk = 0; kk < nk; ++kk) {
    const float* kp = k + ((size_t)(b * nk + kk)) * di + h * dh;
    float s = 0.f;
    for (int d = 0; d < dh; ++d) s += qreg[d] * kp[d];
    s *= scale;
    const float nm = fmaxf(mrun, s);
    const float corr = expf(mrun - nm);
    const float pv = expf(s - nm);
    lsum = lsum * corr + pv;
    const float* vp = v + ((size_t)(b * nk + kk)) * di + h * dh;
    for (int d = 0; d < dh; ++d) acc[d] = acc[d] * corr + pv * vp[d];
    mrun = nm;
  }
  const float inv = 1.f / lsum;
  float* op = out + ((size_t)(b * nq + qi)) * di + h * dh;
  for (int d = 0; d < dh; ++d) op[d] = acc[d] * inv;
}

// Final: LN(keys + a2) then transpose back to (B, C, 16,16,16)
__global__ void final_out_kernel(const float* __restrict__ keys, const float* __restrict__ a2,
                                 const float* __restrict__ w, const float* __restrict__ b,
                                 float* __restrict__ out) {
  const int row = blockIdx.x;
  const int t = row & (LSEQ - 1), bb = row >> 12;
  __shared__ float sb[256];
  float s = 0.f, s2 = 0.f;
  for (int c = threadIdx.x; c < DIMC; c += 256) {
    const float v = keys[(size_t)row * DIMC + c] + a2[(size_t)row * DIMC + c];
    s += v; s2 += v * v;
  }
  sb[threadIdx.x] = s; __syncthreads();
  for (int st = 128; st > 0; st >>= 1) { if (threadIdx.x < st) sb[threadIdx.x] += sb[threadIdx.x + st]; __syncthreads(); }
  const float mean = sb[0] / DIMC; __syncthreads();
  sb[threadIdx.x] = s2; __syncthreads();
  for (int st = 128; st > 0; st >>= 1) { if (threadIdx.x < st) sb[threadIdx.x] += sb[threadIdx.x + st]; __syncthreads(); }
  const float inv = rsqrtf(sb[0] / DIMC - mean * mean + 1e-5f);
  for (int c = threadIdx.x; c < DIMC; c += 256) {
    const float v = keys[(size_t)row * DIMC + c] + a2[(size_t)row * DIMC + c];
    out[((size_t)bb * DIMC + c) * LSEQ + t] = (v - mean) * inv * w[c] + b[c];
  }
}

// ---------------------------------------------------------------------------
extern "C" void kernel_launch(void* const* d_in, const int* in_sizes, int n_in,
                              void* d_out, int out_size, void* d_ws, size_t ws_size,
                              hipStream_t stream) {
  (void)in_sizes; (void)n_in; (void)out_size; (void)ws_size;
  const float* x            = (const float*)d_in[0];
  const float* coords       = (const float*)d_in[1];
  const int*   labels       = (const int*)  d_in[2];
  const float* ln_w         = (const float*)d_in[3];
  const float* ln_b         = (const float*)d_in[4];
  const float* in_proj_w    = (const float*)d_in[5];
  const float* conv_w       = (const float*)d_in[6];
  const float* conv_b       = (const float*)d_in[7];
  const float* dt_bias      = (const float*)d_in[8];
  const float* A_log        = (const float*)d_in[9];
  const float* D_param      = (const float*)d_in[10];
  const float* mamba_norm_w = (const float*)d_in[11];
  const float* out_proj_w   = (const float*)d_in[12];
  const float* pe_gauss     = (const float*)d_in[13];
  const float* point_embeds = (const float*)d_in[14];
  const float* sa_wqkv      = (const float*)d_in[15];
  const float* sa_bqkv      = (const float*)d_in[16];
  const float* sa_wo        = (const float*)d_in[17];
  const float* sa_bo        = (const float*)d_in[18];
  const float* c1_wqkv      = (const float*)d_in[19];
  const float* c1_bqkv      = (const float*)d_in[20];
  const float* c1_wo        = (const float*)d_in[21];
  const float* c1_bo        = (const float*)d_in[22];
  const float* c2_wqkv      = (const float*)d_in[23];
  const float* c2_bqkv      = (const float*)d_in[24];
  const float* c2_wo        = (const float*)d_in[25];
  const float* c2_bo        = (const float*)d_in[26];
  const float* mlp_w1       = (const float*)d_in[27];
  const float* mlp_b1       = (const float*)d_in[28];
  const float* mlp_w2       = (const float*)d_in[29];
  const float* mlp_b2       = (const float*)d_in[30];
  const float* ca_norm_w    = (const float*)d_in[31];
  const float* ca_norm_b    = (const float*)d_in[32];
  float* out = (float*)d_out;

  float* wsp = (float*)d_ws;
  size_t off = 0;
  auto alloc = [&](size_t nf) { float* p = wsp + off; off += nf; return p; };

  float* u     = alloc((size_t)NTOK * DIMC);
  float* zx    = alloc((size_t)NTOK * DPROJ);
  float* xbc   = alloc((size_t)NTOK * CONVD);
  float* dt    = alloc((size_t)NTOK * NHD);
  float* dA    = alloc((size_t)NTOK * NHD);
  float* ym    = alloc((size_t)NTOK * DINNER);
  float* keys  = alloc((size_t)NTOK * DIMC);
  float* kpe   = alloc((size_t)LSEQ * DIMC);
  float* qpe   = alloc(16 * DIMC);
  float* saq   = alloc(16 * DIMC);
  float* sak   = alloc(16 * DIMC);
  float* sav   = alloc(16 * DIMC);
  float* satt  = alloc(16 * DIMC);
  float* sattp = alloc(16 * DIMC);
  float* q0    = alloc(16 * DIMC);
  float* q0p   = alloc(16 * DIMC);
  float* c1q   = alloc(16 * 160);
  float* kvadd = alloc((size_t)NTOK * DIMC);
  float* c1k   = alloc((size_t)NTOK * 160);
  float* c1v   = alloc((size_t)NTOK * 160);
  float* att1  = alloc(16 * 160);
  float* attp  = alloc(16 * DIMC);
  float* q1    = alloc(16 * DIMC);
  float* mlph  = alloc(16 * 2048);
  float* mlpo  = alloc(16 * DIMC);
  float* q2    = alloc(16 * DIMC);
  float* q2p   = alloc(16 * DIMC);
  float* c2q   = alloc((size_t)NTOK * 160);
  float* c2k   = alloc(16 * 160);
  float* c2v   = alloc(16 * 160);
  float* att2  = alloc((size_t)NTOK * 160);
  float* a2    = alloc((size_t)NTOK * DIMC);

  // --- Mamba branch ---
  ln_in_kernel<<<NTOK, 256, 0, stream>>>(x, ln_w, ln_b, u);
  gemm_wmma_lds<<<dim3((DPROJ + 31) / 32, NTOK / 128), 256, 0, stream>>>(u, in_proj_w, nullptr, zx, NTOK, DPROJ, DIMC);
  dtda_kernel<<<(NTOK * NHD + 255) / 256, 256, 0, stream>>>(zx, dt_bias, A_log, dt, dA);
  conv_silu_kernel<<<(NTOK * CONVD + 255) / 256, 256, 0, stream>>>(zx, conv_w, conv_b, xbc);
  scan_kernel<<<16, 128, 0, stream>>>(xbc, dt, dA, D_param, ym);
  gate_kernel<<<(int)(((size_t)NTOK * DINNER + 255) / 256), 256, 0, stream>>>(zx, ym);
  norm_kernel<<<NTOK, 256, 0, stream>>>(ym, nullptr, mamba_norm_w, nullptr, ym, DINNER, 1);
  gemm_wmma_lds<<<dim3(DIMC / 32, NTOK / 128), 256, 0, stream>>>(ym, out_proj_w, nullptr, keys, NTOK, DIMC, DINNER);

  // --- Positional encodings ---
  kpe_kernel<<<(LSEQ * 160 + 255) / 256, 256, 0, stream>>>(pe_gauss, kpe);
  qpe_kernel<<<(16 * 160 + 255) / 256, 256, 0, stream>>>(coords, labels, pe_gauss, point_embeds, qpe);

  // --- Self-attention over the 8 query points ---
  gemm_wmma_f32<<<dim3(DIMC / 16, 1), 32, 0, stream>>>(qpe, sa_wqkv + 0 * DIMC * DIMC, sa_bqkv + 0 * DIMC, saq, 16, DIMC, DIMC, 0);
  gemm_wmma_f32<<<dim3(DIMC / 16, 1), 32, 0, stream>>>(qpe, sa_wqkv + 1 * DIMC * DIMC, sa_bqkv + 1 * DIMC, sak, 16, DIMC, DIMC, 0);
  gemm_wmma_f32<<<dim3(DIMC / 16, 1), 32, 0, stream>>>(qpe, sa_wqkv + 2 * DIMC * DIMC, sa_bqkv + 2 * DIMC, sav, 16, DIMC, DIMC, 0);
  attn_kernel<<<1, 128, 0, stream>>>(saq, sak, sav, satt, 8, 8, 4, 80);
  gemm_wmma_f32<<<dim3(DIMC / 16, 1), 32, 0, stream>>>(satt, sa_wo, sa_bo, sattp, 16, DIMC, DIMC, 0);
  norm_kernel<<<16, 256, 0, stream>>>(sattp, nullptr, ca_norm_w + 0 * DIMC, ca_norm_b + 0 * DIMC, q0, DIMC, 0);

  // --- Cross-attn 1: queries attend to keys ---
  add_bcast_kernel<<<(16 * DIMC + 255) / 256, 256, 0, stream>>>(q0, qpe, q0p, 16 * DIMC, 16 * DIMC);
  add_bcast_kernel<<<(int)(((size_t)NTOK * DIMC + 255) / 256), 256, 0, stream>>>(keys, kpe, kvadd, (size_t)NTOK * DIMC, (size_t)LSEQ * DIMC);
  gemm_wmma_f32<<<dim3(10, 1), 32, 0, stream>>>(q0p, c1_wqkv + 0 * 160 * DIMC, c1_bqkv + 0, c1q, 16, 160, DIMC, 0);
  gemm_wmma_lds<<<dim3(5, NTOK / 128), 256, 0, stream>>>(kvadd, c1_wqkv + 1 * 160 * DIMC, c1_bqkv + 160, c1k, NTOK, 160, DIMC);
  gemm_wmma_lds<<<dim3(5, NTOK / 128), 256, 0, stream>>>(keys, c1_wqkv + 2 * 160 * DIMC, c1_bqkv + 320, c1v, NTOK, 160, DIMC);
  attn_kernel<<<1, 128, 0, stream>>>(c1q, c1k, c1v, att1, 8, LSEQ, 4, 40);
  gemm_wmma_f32<<<dim3(DIMC / 16, 1), 32, 0, stream>>>(att1, c1_wo, c1_bo, attp, 16, DIMC, 160, 0);
  norm_kernel<<<16, 256, 0, stream>>>(q0, attp, ca_norm_w + 1 * DIMC, ca_norm_b + 1 * DIMC, q1, DIMC, 0);

  // --- MLP on queries ---
  gemm_wmma_f32<<<dim3(2048 / 16, 1), 32, 0, stream>>>(q1, mlp_w1, mlp_b1, mlph, 16, 2048, DIMC, 1);
  gemm_wmma_f32<<<dim3(DIMC / 16, 1), 32, 0, stream>>>(mlph, mlp_w2, mlp_b2, mlpo, 16, DIMC, 2048, 0);
  norm_kernel<<<16, 256, 0, stream>>>(q1, mlpo, ca_norm_w + 2 * DIMC, ca_norm_b + 2 * DIMC, q2, DIMC, 0);

  // --- Cross-attn 2: keys attend to queries ---
  add_bcast_kernel<<<(16 * DIMC + 255) / 256, 256, 0, stream>>>(q2, qpe, q2p, 16 * DIMC, 16 * DIMC);
  gemm_wmma_lds<<<dim3(5, NTOK / 128), 256, 0, stream>>>(kvadd, c2_wqkv + 0 * 160 * DIMC, c2_bqkv + 0, c2q, NTOK, 160, DIMC);
  gemm_wmma_f32<<<dim3(10, 1), 32, 0, stream>>>(q2p, c2_wqkv + 1 * 160 * DIMC, c2_bqkv + 160, c2k, 16, 160, DIMC, 0);
  gemm_wmma_f32<<<dim3(10, 1), 32, 0, stream>>>(q2, c2_wqkv + 2 * 160 * DIMC, c2_bqkv + 320, c2v, 16, 160, DIMC, 0);
  attn_kernel<<<(2 * 4 * LSEQ + 127) / 128, 128, 0, stream>>>(c2q, c2k, c2v, att2, LSEQ, 8, 4, 40);
  gemm_wmma_lds<<<dim3(DIMC / 32, NTOK / 128), 256, 0, stream>>>(att2, c2_wo, c2_bo, a2, NTOK, DIMC, 160);

  // --- Final LN + transpose back to (B, C, 16,16,16) ---
  final_out_kernel<<<NTOK, 256, 0, stream>>>(keys, a2, ca_norm_w + 3 * DIMC, ca_norm_b + 3 * DIMC, out);
}